// LayoutLinear_7928509628811
// MI455X (gfx1250) — compile-verified
//
#include <hip/hip_runtime.h>
#include <stdint.h>

#define D_FEAT 256
#define WAVES_PER_BLOCK 8
#define EDGES_PER_WAVE 32

// ---------------------------------------------------------------------------
// Zero-init the output (harness poisons d_out with 0xAA before timing).
// ---------------------------------------------------------------------------
__global__ __launch_bounds__(256) void spmm_zero(float* __restrict__ out, int n) {
    int i4 = (blockIdx.x * blockDim.x + threadIdx.x) * 4;
    if (i4 + 3 < n) {
        *(float4*)(out + i4) = make_float4(0.f, 0.f, 0.f, 0.f);
    } else {
        for (int k = i4; k < n; ++k) out[k] = 0.f;
    }
}

// ---------------------------------------------------------------------------
// SpMM scatter: one wave32 per chunk of 32 edges.
//   Stage (row, col, val) triples into LDS with CDNA5 async global->LDS ops,
//   then per edge: uniform (readfirstlane) row bases, coalesced b128 gathers
//   of the 1KB weight row, scale by val, scatter with global_atomic_add_f32.
// ---------------------------------------------------------------------------
__global__ __launch_bounds__(256) void spmm_scatter(
    const int*   __restrict__ row,
    const int*   __restrict__ col,
    const float* __restrict__ vals,
    const float* __restrict__ weight,
    float*       __restrict__ out,
    int nnz)
{
    __shared__ unsigned srow[WAVES_PER_BLOCK][EDGES_PER_WAVE];
    __shared__ unsigned scol[WAVES_PER_BLOCK][EDGES_PER_WAVE];
    __shared__ float    sval[WAVES_PER_BLOCK][EDGES_PER_WAVE];

    const int w    = threadIdx.x >> 5;   // wave id within block
    const int lane = threadIdx.x & 31;

    const long long chunk = (long long)blockIdx.x * WAVES_PER_BLOCK + w;
    const long long base  = chunk * EDGES_PER_WAVE;
    if (base >= (long long)nnz) return;

    long long ei  = base + lane;
    bool      oob = (ei >= (long long)nnz);
    if (oob) ei = (long long)nnz - 1;     // clamp address; value zeroed below

    // ---- CDNA5 async global->LDS staging (tracked by ASYNCcnt) ----
    // Low 32 bits of a flat shared-aperture address == wave-relative LDS byte
    // offset (aperture base is 4GB-aligned), which is what the ds-addr VGPR
    // of GLOBAL_LOAD_ASYNC_TO_LDS expects.
    unsigned lr = (unsigned)(uintptr_t)&srow[w][lane];
    unsigned lc = (unsigned)(uintptr_t)&scol[w][lane];
    unsigned lv = (unsigned)(uintptr_t)&sval[w][lane];
    uint64_t gr = (uint64_t)(uintptr_t)(row  + ei);
    uint64_t gc = (uint64_t)(uintptr_t)(col  + ei);
    uint64_t gv = (uint64_t)(uintptr_t)(vals + ei);
    asm volatile("global_load_async_to_lds_b32 %0, %1, off" :: "v"(lr), "v"(gr) : "memory");
    asm volatile("global_load_async_to_lds_b32 %0, %1, off" :: "v"(lc), "v"(gc) : "memory");
    asm volatile("global_load_async_to_lds_b32 %0, %1, off" :: "v"(lv), "v"(gv) : "memory");
    asm volatile("s_wait_asynccnt 0x0" ::: "memory");
    if (oob) sval[w][lane] = 0.0f;        // partial tail chunk: contribute 0

    const int PF = 6;                     // prefetch distance in edges
    #pragma unroll 4
    for (int e = 0; e < EDGES_PER_WAVE; ++e) {
        // Prefetch an upcoming gathered row (global_prefetch_b8): 32 lanes x
        // 32B strides cover the full 1KB row.
        if (e + PF < EDGES_PER_WAVE) {
            int cp = __builtin_amdgcn_readfirstlane((int)scol[w][e + PF]);
            __builtin_prefetch(weight + (size_t)cp * D_FEAT + lane * 8, 0, 0);
        }

        // Uniform per-wave scalars from LDS -> SGPR (saddr-form gathers).
        int   c = __builtin_amdgcn_readfirstlane((int)scol[w][e]);
        int   r = __builtin_amdgcn_readfirstlane((int)srow[w][e]);
        float v = sval[w][e];

        // Coalesced gather of weight row c: lane covers features
        // [4*lane, 4*lane+4) and [128+4*lane, 128+4*lane+4).
        const float4* wp = (const float4*)(weight + (size_t)c * D_FEAT);
        float4 a = wp[lane];
        float4 b = wp[lane + 32];

        float a0 = a.x * v, a1 = a.y * v, a2 = a.z * v, a3 = a.w * v;
        float b0 = b.x * v, b1 = b.y * v, b2 = b.z * v, b3 = b.w * v;

        // Scatter-add into out row r with hardware no-return f32 atomics.
        uint64_t oa = (uint64_t)(uintptr_t)(out + (size_t)r * D_FEAT + 4 * lane);
        uint64_t ob = oa + 512;           // +128 features
        asm volatile(
            "global_atomic_add_f32 %0, %2, off\n\t"
            "global_atomic_add_f32 %0, %3, off offset:4\n\t"
            "global_atomic_add_f32 %0, %4, off offset:8\n\t"
            "global_atomic_add_f32 %0, %5, off offset:12\n\t"
            "global_atomic_add_f32 %1, %6, off\n\t"
            "global_atomic_add_f32 %1, %7, off offset:4\n\t"
            "global_atomic_add_f32 %1, %8, off offset:8\n\t"
            "global_atomic_add_f32 %1, %9, off offset:12"
            :: "v"(oa), "v"(ob),
               "v"(a0), "v"(a1), "v"(a2), "v"(a3),
               "v"(b0), "v"(b1), "v"(b2), "v"(b3)
            : "memory");
    }
}

// ---------------------------------------------------------------------------
// Host-side launcher. Inputs (setup_inputs order): row(i32), col(i32),
// vals(f32), weight(f32 [N, 256]). Output: f32 [N, 256].
// ---------------------------------------------------------------------------
extern "C" void kernel_launch(void* const* d_in, const int* in_sizes, int n_in,
                              void* d_out, int out_size, void* d_ws, size_t ws_size,
                              hipStream_t stream) {
    const int*   row    = (const int*)d_in[0];
    const int*   col    = (const int*)d_in[1];
    const float* vals   = (const float*)d_in[2];
    const float* weight = (const float*)d_in[3];
    float*       out    = (float*)d_out;
    const int    nnz    = in_sizes[0];

    // Zero the output.
    int n4 = (out_size + 3) / 4;
    int zblocks = (n4 + 255) / 256;
    spmm_zero<<<zblocks, 256, 0, stream>>>(out, out_size);

    // Scatter kernel: one wave per 32-edge chunk, 8 waves per block.
    long long nchunks = ((long long)nnz + EDGES_PER_WAVE - 1) / EDGES_PER_WAVE;
    int sblocks = (int)((nchunks + WAVES_PER_BLOCK - 1) / WAVES_PER_BLOCK);
    spmm_scatter<<<sblocks, 256, 0, stream>>>(row, col, vals, weight, out, nnz);
}